// SetAbstraction_61787399520989
// MI455X (gfx1250) — compile-verified
//
#include <hip/hip_runtime.h>
#include <hip/hip_bf16.h>

#define B_      4
#define N_      8192
#define C_      96
#define NPOINT_ 2048
#define NSAMPLE_ 32
#define RADIUS2 0.01f

typedef __attribute__((ext_vector_type(2))) float v2f;
typedef __attribute__((ext_vector_type(8))) float v8f;

// ---------------------------------------------------------------------------
// Kernel 1: farthest point sampling. One 1024-thread block per batch.
// Points staged in LDS (SoA) for O(1) lookup of the last pick; each thread's
// 8 points + running min-distances live in registers, so the 2047-step serial
// chain costs only 3 LDS reads + ~70 VALU + a shfl argmax tree per step.
// ---------------------------------------------------------------------------
__global__ __launch_bounds__(1024) void fps_kernel(const float* __restrict__ p,
                                                   float* __restrict__ out_newp,
                                                   int* __restrict__ samp)
{
    extern __shared__ float smem[];
    float* px = smem;               // N_ floats
    float* py = smem + N_;
    float* pz = smem + 2 * N_;
    float* wv = smem + 3 * N_;      // 32 wave maxima
    int*   wi = (int*)(smem + 3 * N_ + 32);
    int*   sLast = (int*)(smem + 3 * N_ + 64);

    const int b    = blockIdx.x;
    const int tid  = threadIdx.x;
    const int lane = tid & 31;
    const int wave = tid >> 5;
    const float* pb = p + (size_t)b * N_ * 3;

    for (int i = tid; i < N_; i += 1024) {
        px[i] = pb[i * 3 + 0];
        py[i] = pb[i * 3 + 1];
        pz[i] = pb[i * 3 + 2];
    }
    __syncthreads();

    // thread-private copy of its 8 strided points + running distances
    float rx[8], ry[8], rz[8], rd[8];
#pragma unroll
    for (int r = 0; r < 8; ++r) {
        int j = tid + r * 1024;
        rx[r] = px[j]; ry[r] = py[j]; rz[r] = pz[j];
        rd[r] = 1e10f;
    }

    int last = 0;
    if (tid == 0) {
        samp[b * NPOINT_] = 0;
        out_newp[(size_t)(b * NPOINT_) * 3 + 0] = px[0];
        out_newp[(size_t)(b * NPOINT_) * 3 + 1] = py[0];
        out_newp[(size_t)(b * NPOINT_) * 3 + 2] = pz[0];
    }

    for (int it = 1; it < NPOINT_; ++it) {
        float lx = px[last], ly = py[last], lz = pz[last];
        float bv = -1.0f;
        int   bi = 0x7fffffff;
#pragma unroll
        for (int r = 0; r < 8; ++r) {
            int j = tid + r * 1024;
            float dx = rx[r] - lx, dy = ry[r] - ly, dz = rz[r] - lz;
            // match jnp: ((x^2 + y^2) + z^2), no FMA contraction
            float d = __fadd_rn(__fadd_rn(__fmul_rn(dx, dx), __fmul_rn(dy, dy)),
                                __fmul_rn(dz, dz));
            float dmin = fminf(rd[r], d);
            rd[r] = dmin;
            if (dmin > bv || (dmin == bv && j < bi)) { bv = dmin; bi = j; }
        }
        // wave32 argmax (ties -> lower index, as jnp.argmax)
#pragma unroll
        for (int off = 16; off > 0; off >>= 1) {
            float ov = __shfl_down(bv, off);
            int   oi = __shfl_down(bi, off);
            if (ov > bv || (ov == bv && oi < bi)) { bv = ov; bi = oi; }
        }
        if (lane == 0) { wv[wave] = bv; wi[wave] = bi; }
        __syncthreads();
        if (wave == 0) {
            float v2 = wv[lane];
            int   i2 = wi[lane];
#pragma unroll
            for (int off = 16; off > 0; off >>= 1) {
                float ov = __shfl_down(v2, off);
                int   oi = __shfl_down(i2, off);
                if (ov > v2 || (ov == v2 && oi < i2)) { v2 = ov; i2 = oi; }
            }
            if (lane == 0) {
                *sLast = i2;
                samp[b * NPOINT_ + it] = i2;
                out_newp[(size_t)(b * NPOINT_ + it) * 3 + 0] = px[i2];
                out_newp[(size_t)(b * NPOINT_ + it) * 3 + 1] = py[i2];
                out_newp[(size_t)(b * NPOINT_ + it) * 3 + 2] = pz[i2];
            }
        }
        __syncthreads();
        last = *sLast;
    }
}

// ---------------------------------------------------------------------------
// Kernel 2: fused ball query (WMMA f32 16x16x4 distance tiles) + positional
// embedding + aggregation. All 8 waves of a block share one batch, so the
// batch's 8192 points are staged ONCE in LDS as float4(x,y,z,|p|^2) (128 KB of
// the 320 KB WGP LDS; 2 blocks/WGP still fit). The tile loop is 2x unrolled:
// 2 ds_load_b128 feed 2 back-to-back v_wmma_f32_16x16x4_f32. Hits appended in
// ascending point index (low 16 ballot bits = acc row v, high 16 = row v+8).
// Phase 2: lane = channel-column t (f = t&15, sin/cos fixed per lane), so the
// max/mean over 32 samples is straight register accumulation — no wave
// reductions; cos folds into v_sin via a +0.25 revolution offset.
// ---------------------------------------------------------------------------
__global__ __launch_bounds__(256) void sa_kernel(const float* __restrict__ p,
                                                 const float* __restrict__ x,
                                                 const float* __restrict__ newp,
                                                 float* __restrict__ feat)
{
    extern __shared__ float4 spt[];               // N_ float4 = 128 KB
    int* nb  = (int*)(spt + N_);                  // [8][16][NSAMPLE_]
    int* cnt = nb + 8 * 16 * NSAMPLE_;            // [8][16]

    const int tid  = threadIdx.x;
    const int lane = tid & 31;
    const int w    = tid >> 5;
    const int gg   = blockIdx.x * 8 + w;          // group of 16 centers
    const int b    = gg >> 7;                     // 128 groups per batch
    const int gbase = (gg & 127) * 16;
    const int m    = lane & 15;
    const bool lo  = (lane < 16);

    const float* pb = p + (size_t)b * N_ * 3;

    // stage batch points + precomputed norms into LDS (cooperative)
    for (int i = tid; i < N_; i += 256) {
        float qx = pb[i * 3 + 0];
        float qy = pb[i * 3 + 1];
        float qz = pb[i * 3 + 2];
        spt[i] = make_float4(qx, qy, qz, qx * qx + qy * qy + qz * qz);
    }
    if (lane < 16) cnt[w * 16 + lane] = 0;
    __syncthreads();

    const int cidx = gbase + m;
    const float cx = newp[(size_t)(b * NPOINT_ + cidx) * 3 + 0];
    const float cy = newp[(size_t)(b * NPOINT_ + cidx) * 3 + 1];
    const float cz = newp[(size_t)(b * NPOINT_ + cidx) * 3 + 2];
    const float cn = cx * cx + cy * cy + cz * cz;

    // A-matrix 16x4 f32: lanes 0-15 hold K=0,1; lanes 16-31 hold K=2,3
    v2f A;
    A.x = lo ? cx : cz;
    A.y = lo ? cy : 0.0f;

    // (|c|^2 - r^2) for the row this lane-half sees in accumulator VGPR v
    float cns[8];
#pragma unroll
    for (int v = 0; v < 8; ++v)
        cns[v] = __shfl(cn, v + (lo ? 0 : 8)) - RADIUS2;

    // hit test + in-order append for one 16x16 tile
    auto process = [&](const v8f& acc, float pn, int base) {
#pragma unroll
        for (int v = 0; v < 8; ++v) {
            bool hit = (cns[v] + pn) < 2.0f * acc[v];
            unsigned mask = (unsigned)__ballot(hit);
            if (mask) {
                if (lane == 0) {                  // row v, cols base..base+15
                    unsigned mlo = mask & 0xffffu;
                    int c0 = cnt[w * 16 + v];
                    while (mlo && c0 < NSAMPLE_) {
                        int bit = __ffs(mlo) - 1;
                        nb[(w * 16 + v) * NSAMPLE_ + c0++] = base + bit;
                        mlo &= mlo - 1;
                    }
                    cnt[w * 16 + v] = c0;
                } else if (lane == 16) {          // row v+8
                    unsigned mhi = mask >> 16;
                    int row = v + 8;
                    int c0 = cnt[w * 16 + row];
                    while (mhi && c0 < NSAMPLE_) {
                        int bit = __ffs(mhi) - 1;
                        nb[(w * 16 + row) * NSAMPLE_ + c0++] = base + bit;
                        mhi &= mhi - 1;
                    }
                    cnt[w * 16 + row] = c0;
                }
            }
        }
    };

    for (int n0 = 0; n0 < N_; n0 += 32) {
        float4 t0 = spt[n0 + m];
        float4 t1 = spt[n0 + 16 + m];
        v2f B0, B1;
        B0.x = lo ? t0.x : t0.z;  B0.y = lo ? t0.y : 0.0f;
        B1.x = lo ? t1.x : t1.z;  B1.y = lo ? t1.y : 0.0f;
        v8f acc0 = {}, acc1 = {};
        acc0 = __builtin_amdgcn_wmma_f32_16x16x4_f32(
            false, A, false, B0, (short)0, acc0, false, false);
        acc1 = __builtin_amdgcn_wmma_f32_16x16x4_f32(
            false, A, false, B1, (short)0, acc1, false, false);
        process(acc0, t0.w, n0);
        process(acc1, t1.w, n0 + 16);
    }
    __syncthreads();

    // ---- Phase 2: lane = channel-column t; owns channels {t, 32+t, 64+t} ---
    // pe(c = d*32 + t) = sin(50*dp[d]*500^(-f/16))        for t < 16 (f = t)
    //                  = cos(50*dp[d]*500^(-f/16))        for t >= 16 (f = t-16)
    // v_sin_f32 takes revolutions: rev = ang/(2pi); cos = sin(rev + 0.25).
    const int   f    = lane & 15;
    const float offs = lo ? 0.0f : 0.25f;
    const float myrev = 50.0f * __powf(500.0f, -(float)f * (1.0f / 16.0f))
                        * 0.15915494309189535f;   // fold 1/(2pi)

    const float* xb = x + (size_t)b * C_ * N_;
    float* fb = feat + (size_t)b * C_ * NPOINT_;

    for (int mm = 0; mm < 16; ++mm) {
        int count = cnt[w * 16 + mm];
        int base  = (w * 16 + mm) * NSAMPLE_;
        int idx0  = (count > 0) ? nb[base] : 0;            // pad-with-first rule
        int myidx = (lane < count) ? nb[base + lane] : idx0;  // sample for lane=k
        float ccx = __shfl(cx, mm);
        float ccy = __shfl(cy, mm);
        float ccz = __shfl(cz, mm);
        float4 q = spt[myidx];
        float d0 = q.x - ccx;
        float d1 = q.y - ccy;
        float d2 = q.z - ccz;

        float mx0 = -1e30f, mx1 = -1e30f, mx2 = -1e30f;
        float sm0 = 0.0f,   sm1 = 0.0f,   sm2 = 0.0f;

        for (int k = 0; k < NSAMPLE_; ++k) {
            int   ik = __shfl(myidx, k);                   // uniform broadcast
            float e0 = __shfl(d0, k);
            float e1 = __shfl(d1, k);
            float e2 = __shfl(d2, k);
            float p0 = __builtin_amdgcn_sinf(__fmaf_rn(e0, myrev, offs));
            float p1 = __builtin_amdgcn_sinf(__fmaf_rn(e1, myrev, offs));
            float p2 = __builtin_amdgcn_sinf(__fmaf_rn(e2, myrev, offs));
            float x0 = xb[(size_t)lane * N_ + ik];
            float x1 = xb[(size_t)(lane + 32) * N_ + ik];
            float x2 = xb[(size_t)(lane + 64) * N_ + ik];
            float g0 = (x0 + p0) * p0;
            float g1 = (x1 + p1) * p1;
            float g2 = (x2 + p2) * p2;
            mx0 = fmaxf(mx0, g0);  sm0 += g0;
            mx1 = fmaxf(mx1, g1);  sm1 += g1;
            mx2 = fmaxf(mx2, g2);  sm2 += g2;
        }
        int cglob = gbase + mm;
        fb[(size_t)lane * NPOINT_ + cglob]        = mx0 + sm0 * (1.0f / 32.0f);
        fb[(size_t)(lane + 32) * NPOINT_ + cglob] = mx1 + sm1 * (1.0f / 32.0f);
        fb[(size_t)(lane + 64) * NPOINT_ + cglob] = mx2 + sm2 * (1.0f / 32.0f);
    }
}

extern "C" void kernel_launch(void* const* d_in, const int* in_sizes, int n_in,
                              void* d_out, int out_size, void* d_ws, size_t ws_size,
                              hipStream_t stream) {
    (void)in_sizes; (void)n_in; (void)out_size; (void)ws_size;
    const float* p = (const float*)d_in[0];   // (B, N, 3)
    const float* x = (const float*)d_in[1];   // (B, C, N)
    float* out  = (float*)d_out;
    float* newp = out;                        // (B, NPOINT, 3)
    float* feat = out + (size_t)B_ * NPOINT_ * 3;   // (B, C, NPOINT)
    int* samp = (int*)d_ws;                   // (B, NPOINT) scratch

    size_t smem_fps = (size_t)(3 * N_ + 80) * sizeof(float);  // ~96.3 KB
    fps_kernel<<<B_, 1024, smem_fps, stream>>>(p, newp, samp);

    // 128 KB points + 16.5 KB neighbor lists -> 2 blocks per 320 KB WGP
    size_t smem_sa = (size_t)N_ * sizeof(float4)
                   + (size_t)(8 * 16 * NSAMPLE_ + 8 * 16) * sizeof(int);
    const int groups = (B_ * NPOINT_) / 16;   // 512 waves of 16 centers
    sa_kernel<<<groups / 8, 256, smem_sa, stream>>>(p, x, newp, feat);
}